// HyperGCN_11922829213805
// MI455X (gfx1250) — compile-verified
//
#include <hip/hip_runtime.h>

// ---------------- problem constants (fixed by the reference) ----------------
#define N_NODES 100000
#define N_EDGES 100000
#define C_HID   128
#define C_OUT   64

typedef __attribute__((ext_vector_type(2))) float v2f;
typedef __attribute__((ext_vector_type(8))) float v8f;

// ============================================================================
// fp32 WMMA GEMM:  Cout[M, Nc] = X[M, K] @ W[Nc, K]^T  (+ optional bias[Nc])
// One wave (32 lanes) computes one 16x16 output tile via V_WMMA_F32_16X16X4_F32.
// A(16x4) lane layout: lanes 0-15 M=idx hold K={0,1}; lanes 16-31 hold K={2,3}.
// B(4x16) is the mirror: lane idx = N column, halves split K pairs.
// C/D(16x16): VGPR r, lanes 0-15 -> M=r, lanes 16-31 -> M=r+8; N = lane&15.
// K is always 128 here; the loop is fully unrolled -> 32 v_wmma per tile.
// ============================================================================
__global__ __launch_bounds__(32)
void gemm_xwt_wmma(const float* __restrict__ X, const float* __restrict__ W,
                   const float* __restrict__ bias, float* __restrict__ Cout,
                   int Nc, int K) {
  const int mt   = blockIdx.x * 16;
  const int nt   = blockIdx.y * 16;
  const int lane = threadIdx.x;
  const int half = lane >> 4;
  const int idx  = lane & 15;

  const float* Xrow = X + (size_t)(mt + idx) * K + 2 * half;  // A: row mt+idx
  const float* Wrow = W + (size_t)(nt + idx) * K + 2 * half;  // B^T: W row nt+idx

  v8f acc = {0.f, 0.f, 0.f, 0.f, 0.f, 0.f, 0.f, 0.f};
#pragma unroll
  for (int k = 0; k < 128; k += 4) {
    v2f a = *(const v2f*)(Xrow + k);   // 8B-aligned global_load_b64
    v2f b = *(const v2f*)(Wrow + k);
    acc = __builtin_amdgcn_wmma_f32_16x16x4_f32(
        /*neg_a=*/false, a, /*neg_b=*/false, b,
        /*c_mod=*/(short)0, acc, /*reuse_a=*/false, /*reuse_b=*/false);
  }

  const float bv = bias ? bias[nt + idx] : 0.0f;
#pragma unroll
  for (int r = 0; r < 8; ++r)
    Cout[(size_t)(mt + r + 8 * half) * Nc + (nt + idx)] = acc[r] + bv;
}

// ---------------------------------------------------------------------------
__global__ __launch_bounds__(256)
void zero_f32(float* __restrict__ p, long n) {
  long i = (long)blockIdx.x * blockDim.x + threadIdx.x;
  const long stride = (long)gridDim.x * blockDim.x;
  for (; i < n; i += stride) p[i] = 0.0f;
}

__global__ __launch_bounds__(256)
void degree_count(const int* __restrict__ nidx, const int* __restrict__ eidx,
                  float* __restrict__ Ddeg, float* __restrict__ Bdeg, int nnz) {
  const int i = blockIdx.x * blockDim.x + threadIdx.x;
  if (i < nnz) {
    atomicAdd(&Bdeg[eidx[i]], 1.0f);
    atomicAdd(&Ddeg[nidx[i]], 1.0f);
  }
}

__global__ __launch_bounds__(256)
void invert_deg(float* __restrict__ p, int n) {
  const int i = blockIdx.x * blockDim.x + threadIdx.x;
  if (i < n) {
    const float v = p[i];
    p[i] = (v > 0.0f) ? (1.0f / v) : 0.0f;
  }
}

// node -> hyperedge: dst[e,:] += src[n,:]   (one wave per nz, float4 per lane)
__global__ __launch_bounds__(256)
void scatter_n2e(const float* __restrict__ src, const int* __restrict__ nidx,
                 const int* __restrict__ eidx, float* __restrict__ dst, int nnz) {
  const long t  = (long)blockIdx.x * blockDim.x + threadIdx.x;
  const long nz = t >> 5;
  if (nz >= nnz) return;
  const int c = (int)(t & 31) << 2;                      // channel group of 4
  const int n = nidx[nz];
  const int e = eidx[nz];
  const float4 v = *(const float4*)(src + (size_t)n * C_HID + c);
  float* d = dst + (size_t)e * C_HID + c;
  atomicAdd(d + 0, v.x); atomicAdd(d + 1, v.y);
  atomicAdd(d + 2, v.z); atomicAdd(d + 3, v.w);
}

// hyperedge -> node with fused B^{-1} scale: dst[n,:] += Binv[e] * src[e,:]
__global__ __launch_bounds__(256)
void scatter_e2n(const float* __restrict__ src, const float* __restrict__ binv,
                 const int* __restrict__ nidx, const int* __restrict__ eidx,
                 float* __restrict__ dst, int nnz) {
  const long t  = (long)blockIdx.x * blockDim.x + threadIdx.x;
  const long nz = t >> 5;
  if (nz >= nnz) return;
  const int c = (int)(t & 31) << 2;
  const int n = nidx[nz];
  const int e = eidx[nz];
  const float s = binv[e];
  const float4 v = *(const float4*)(src + (size_t)e * C_HID + c);
  float* d = dst + (size_t)n * C_HID + c;
  atomicAdd(d + 0, v.x * s); atomicAdd(d + 1, v.y * s);
  atomicAdd(d + 2, v.z * s); atomicAdd(d + 3, v.w * s);
}

// h[n,c] = relu(h[n,c] * Dinv[n] + bias[c])
__global__ __launch_bounds__(256)
void finalize_relu(float* __restrict__ h, const float* __restrict__ dinv,
                   const float* __restrict__ bias, int rows) {
  const long i = (long)blockIdx.x * blockDim.x + threadIdx.x;
  if (i >= (long)rows * C_HID) return;
  const int row = (int)(i >> 7);   // C_HID == 128
  const int c   = (int)(i & 127);
  const float v = h[i] * dinv[row] + bias[c];
  h[i] = (v > 0.0f) ? v : 0.0f;
}

// ============================================================================
extern "C" void kernel_launch(void* const* d_in, const int* in_sizes, int n_in,
                              void* d_out, int out_size, void* d_ws, size_t ws_size,
                              hipStream_t stream) {
  const float* x  = (const float*)d_in[0];   // [N, 128]
  const int*   ei = (const int*)  d_in[1];   // [2, NNZ]
  const float* W1 = (const float*)d_in[2];   // [128, 128]
  const float* b1 = (const float*)d_in[3];   // [128]
  const float* W2 = (const float*)d_in[4];   // [128, 128]
  const float* b2 = (const float*)d_in[5];   // [128]
  const float* Wl = (const float*)d_in[6];   // [64, 128]
  const float* bl = (const float*)d_in[7];   // [64]
  float* out = (float*)d_out;                // [N, 64]

  const int N = N_NODES, E = N_EDGES;
  const int nnz = in_sizes[1] / 2;
  const int* nidx = ei;            // row 0: node indices
  const int* eidx = ei + nnz;      // row 1: hyperedge indices

  // ---- workspace layout (floats) ----
  float* Binv = (float*)d_ws;                 // [E]
  float* Dinv = Binv + E;                     // [N]  (contiguous with Binv)
  float* bufA = Dinv + N;                     // [N, 128]  xl
  float* bufB = bufA + (size_t)N * C_HID;     // [E, 128]  m
  float* bufC = bufB + (size_t)E * C_HID;     // [N, 128]  h

  const int  deg_blocks  = (E + N + 255) / 256;
  const int  nz_blocks   = (nnz + 255) / 256;
  const long scat_thr    = (long)nnz * 32;
  const int  scat_blocks = (int)((scat_thr + 255) / 256);
  const int  fin_blocks  = (int)(((long)N * C_HID + 255) / 256);
  const dim3 gemm_grid(N / 16, C_HID / 16);   // 6250 x 8
  const dim3 lin_grid (N / 16, C_OUT / 16);   // 6250 x 4

  // ---- degrees: B^{-1}, D^{-1} (shared by both layers) ----
  zero_f32<<<1024, 256, 0, stream>>>(Binv, (long)(E + N));
  degree_count<<<nz_blocks, 256, 0, stream>>>(nidx, eidx, Dinv, Binv, nnz);
  invert_deg<<<deg_blocks, 256, 0, stream>>>(Binv, E + N);

  // ---- layer 1 ----
  gemm_xwt_wmma<<<gemm_grid, 32, 0, stream>>>(x, W1, nullptr, bufA, C_HID, C_HID);
  zero_f32<<<2048, 256, 0, stream>>>(bufB, (long)E * C_HID);
  scatter_n2e<<<scat_blocks, 256, 0, stream>>>(bufA, nidx, eidx, bufB, nnz);
  zero_f32<<<2048, 256, 0, stream>>>(bufC, (long)N * C_HID);
  scatter_e2n<<<scat_blocks, 256, 0, stream>>>(bufB, Binv, nidx, eidx, bufC, nnz);
  finalize_relu<<<fin_blocks, 256, 0, stream>>>(bufC, Dinv, b1, N);

  // ---- layer 2 ----
  gemm_xwt_wmma<<<gemm_grid, 32, 0, stream>>>(bufC, W2, nullptr, bufA, C_HID, C_HID);
  zero_f32<<<2048, 256, 0, stream>>>(bufB, (long)E * C_HID);
  scatter_n2e<<<scat_blocks, 256, 0, stream>>>(bufA, nidx, eidx, bufB, nnz);
  zero_f32<<<2048, 256, 0, stream>>>(bufC, (long)N * C_HID);
  scatter_e2n<<<scat_blocks, 256, 0, stream>>>(bufB, Binv, nidx, eidx, bufC, nnz);
  finalize_relu<<<fin_blocks, 256, 0, stream>>>(bufC, Dinv, b2, N);

  // ---- output linear (bias fused into GEMM epilogue) ----
  gemm_xwt_wmma<<<lin_grid, 32, 0, stream>>>(bufC, Wl, bl, out, C_OUT, C_HID);
}